// TPSRoIExtractor_41901700940191
// MI455X (gfx1250) — compile-verified
//
#include <hip/hip_runtime.h>
#include <hip/hip_bf16.h>
#include <math.h>

typedef __attribute__((ext_vector_type(2))) float v2f;
typedef __attribute__((ext_vector_type(8))) float v8f;
typedef __attribute__((ext_vector_type(4))) unsigned int u32x4;
typedef __attribute__((ext_vector_type(4))) int i32x4;
typedef __attribute__((ext_vector_type(8))) int i32x8;

#define F_PTS  14
#define NCTRL  17            // F_PTS + 3
#define NPTS   256           // OUT_H * OUT_W
#define PHPAD  20            // P_HAT K padded 17 -> 20 (5 WMMA K-steps of 4)
#define PH_TOT (NPTS*PHPAD)  // 5120 floats = 20 KB
#define TPS_EPS 1e-6f

// problem dims (from reference setup_inputs)
#define BD   2
#define CD   256
#define HD   160
#define WD   256
#define HWD  (HD*WD)
#define CH_SPLIT 4
#define CH_PER   (CD/CH_SPLIT)

// ---------------------------------------------------------------------------
// Setup kernel: build INV_DELTA_C (17x17, Gauss-Jordan) and P_HAT (256x20,
// zero-padded K) into workspace. Deterministic; re-runs identically per launch.
// ---------------------------------------------------------------------------
__global__ __launch_bounds__(256) void tps_setup_kernel(float* __restrict__ inv_dc,
                                                        float* __restrict__ p_hat) {
  __shared__ float M[NCTRL][2*NCTRL];   // augmented [delta_C | I]
  const int t = threadIdx.x;

  // control points C: linspace(-1,1,7) x {-1,+1}
  float Cx[F_PTS], Cy[F_PTS];
  #pragma unroll
  for (int i = 0; i < F_PTS; ++i) {
    Cx[i] = -1.0f + (float)(i % 7) * (2.0f / 6.0f);
    Cy[i] = (i < 7) ? -1.0f : 1.0f;
  }

  if (t == 0) {
    for (int r = 0; r < F_PTS; ++r) {
      M[r][0] = 1.0f; M[r][1] = Cx[r]; M[r][2] = Cy[r];
      for (int c = 0; c < F_PTS; ++c) {
        float dx = Cx[r]-Cx[c], dy = Cy[r]-Cy[c];
        float d  = sqrtf(dx*dx + dy*dy);
        if (r == c) d = 1.0f;              // fill_diagonal -> 1^2*log(1)=0
        M[r][3+c] = d*d*logf(d);
      }
    }
    for (int c = 0; c < 3; ++c) { M[14][c]=0.f; M[15][c]=0.f; M[16][c]=0.f; }
    for (int c = 0; c < F_PTS; ++c) {
      M[14][3+c] = Cx[c]; M[15][3+c] = Cy[c]; M[16][3+c] = 1.0f;
    }
    for (int r = 0; r < NCTRL; ++r)
      for (int c = 0; c < NCTRL; ++c)
        M[r][NCTRL+c] = (r == c) ? 1.0f : 0.0f;
    // Gauss-Jordan with partial pivoting
    for (int col = 0; col < NCTRL; ++col) {
      int piv = col; float best = fabsf(M[col][col]);
      for (int r = col+1; r < NCTRL; ++r) {
        float a = fabsf(M[r][col]);
        if (a > best) { best = a; piv = r; }
      }
      if (piv != col)
        for (int c = 0; c < 2*NCTRL; ++c) { float tmp = M[col][c]; M[col][c] = M[piv][c]; M[piv][c] = tmp; }
      float inv = 1.0f / M[col][col];
      for (int c = 0; c < 2*NCTRL; ++c) M[col][c] *= inv;
      for (int r = 0; r < NCTRL; ++r) {
        if (r == col) continue;
        float f = M[r][col];
        for (int c = 0; c < 2*NCTRL; ++c) M[r][c] -= f * M[col][c];
      }
    }
    for (int r = 0; r < NCTRL; ++r)
      for (int c = 0; c < NCTRL; ++c)
        inv_dc[r*NCTRL + c] = M[r][NCTRL + c];
  }

  // P_HAT row n = t : [1, Px, Py, rbf_0..13, 0,0,0]
  const int i = t >> 5, j = t & 31;
  const float Px = (2.0f*(float)j + 1.0f - 32.0f) / 32.0f;   // OUT_W = 32
  const float Py = (2.0f*(float)i + 1.0f - 8.0f) / 8.0f;     // OUT_H = 8
  float* row = p_hat + t * PHPAD;
  row[0] = 1.0f; row[1] = Px; row[2] = Py;
  #pragma unroll
  for (int f = 0; f < F_PTS; ++f) {
    float dx = Px - Cx[f], dy = Py - Cy[f];
    float d  = sqrtf(dx*dx + dy*dy);
    row[3+f] = d*d*logf(d + TPS_EPS);
  }
  row[17] = 0.f; row[18] = 0.f; row[19] = 0.f;
}

// ---------------------------------------------------------------------------
// Main kernel: grid (K, CH_SPLIT), 256 threads (8 wave32 / block).
//  Phase 0: TDM tensor_load_to_lds stages P_HAT (20KB) into LDS (TENSORcnt),
//           overlapped with Phase 1.
//  Phase 1: bbox + Cp + T = INV_DELTA_C @ Cp       (tiny, lanes 0..19)
//  Phase 2: Pp = P_HAT @ T  via V_WMMA_F32_16X16X4_F32 (2 M-tiles / wave,
//           A from ds_load_b64, B fragments built once, branchlessly)
//  Phase 3: per-lane bilinear gather over 64 channels (coalesced b32 stores)
// ---------------------------------------------------------------------------
__global__ __launch_bounds__(256) void tps_main_kernel(
    const float* __restrict__ feats, const float* __restrict__ points,
    const int* __restrict__ bidx, const float* __restrict__ inv_dc,
    const float* __restrict__ p_hat, float* __restrict__ out) {
  const int k = blockIdx.x;
  const int t = threadIdx.x;

  __shared__ __align__(16) float sPH[PH_TOT]; // LDS copy of P_HAT (256x20)
  __shared__ float sMeta[4];                  // x1f, y1f, wc, hc
  __shared__ float sCpX[NCTRL], sCpY[NCTRL];
  __shared__ float sT[PHPAD][2];              // T padded to 20 rows
  __shared__ float sPpx[NPTS], sPpy[NPTS];

  // ---- Phase 0: stage P_HAT into LDS (TDM if available) ----
#if __has_builtin(__builtin_amdgcn_tensor_load_to_lds)
  if (t == 0) {
    // Tensor DMA descriptor (D#), 1-D tile of 5120 x 4B elements.
    const unsigned long long ga = (unsigned long long)(uintptr_t)p_hat;
    const unsigned int lds_off  = (unsigned int)(uintptr_t)&sPH[0]; // low 32b = LDS offset
    u32x4 g0 = { 1u,                                        // count=1 (user, valid)
                 lds_off,                                   // lds_addr (bytes)
                 (unsigned int)(ga & 0xffffffffu),          // global_addr[31:0]
                 (unsigned int)((ga >> 32) & 0x01ffffffu)   // global_addr[56:32]
                   | (2u << 30) };                          // type=2 ("image")
    i32x8 g1 = { (int)(2u << 16),          // wg_mask=0, data_size=2 (4B)
                 (int)((PH_TOT & 0xffff) << 16),   // tensor_dim0[15:0] @ bits 63:48
                 (int)(1u << 16),          // tensor_dim0[31:16]=0, tensor_dim1=1
                 (int)((PH_TOT & 0xffff) << 16),   // tile_dim0 @ bits 127:112
                 1,                        // tile_dim1=1, tile_dim2=0
                 PH_TOT,                   // tensor_dim0_stride[31:0]
                 0, 0 };                   // stride hi / dim1 stride = 0
    i32x4 gz = { 0, 0, 0, 0 };
#if defined(__clang_major__) && __clang_major__ >= 23
    i32x8 gz8 = { 0, 0, 0, 0, 0, 0, 0, 0 };
    __builtin_amdgcn_tensor_load_to_lds(g0, g1, gz, gz, gz8, 0);
#else
    __builtin_amdgcn_tensor_load_to_lds(g0, g1, gz, gz, 0);
#endif
  }
#else
  for (int i = t; i < PH_TOT; i += 256) sPH[i] = p_hat[i];
#endif

  const float* pk = points + k * F_PTS * 2;

  // ---- Phase 1a: bounding box (serial on lane 0; 14 points) ----
  if (t == 0) {
    float mnx = 1e30f, mxx = -1e30f, mny = 1e30f, mxy = -1e30f;
    for (int f = 0; f < F_PTS; ++f) {
      float cx = fminf(fmaxf(pk[2*f+0] * 0.25f, 0.0f), (float)WD);
      float cy = fminf(fmaxf(pk[2*f+1] * 0.25f, 0.0f), (float)HD);
      mnx = fminf(mnx, cx); mxx = fmaxf(mxx, cx);
      mny = fminf(mny, cy); mxy = fmaxf(mxy, cy);
    }
    int x1 = (int)floorf(mnx), x2 = (int)floorf(mxx) + 1;
    int y1 = (int)floorf(mny), y2 = (int)floorf(mxy) + 1;
    if ((x2 > WD) || (x1 == WD)) { x1 = WD - 1; x2 = WD; }
    if ((y2 > HD) && (y1 == HD)) { y1 = HD - 1; y2 = HD; }
    sMeta[0] = (float)x1; sMeta[1] = (float)y1;
    sMeta[2] = (float)(x2 - x1); sMeta[3] = (float)(y2 - y1);
  }
  __syncthreads();

  // ---- Phase 1b: Cp (17x2, rows 14..16 = 0) ----
  if (t < NCTRL) {
    float vx = 0.f, vy = 0.f;
    if (t < F_PTS) {
      vx = 2.0f * (pk[2*t+0] * 0.25f - sMeta[0]) / sMeta[2] - 1.0f;
      vy = 2.0f * (pk[2*t+1] * 0.25f - sMeta[1]) / sMeta[3] - 1.0f;
    }
    sCpX[t] = vx; sCpY[t] = vy;
  }
  __syncthreads();

  // ---- Phase 1c: T = INV_DELTA_C @ Cp  (only Cp rows 0..13 nonzero) ----
  if (t < PHPAD) {
    float t0 = 0.f, t1 = 0.f;
    if (t < NCTRL) {
      const float* r = inv_dc + t * NCTRL;
      #pragma unroll
      for (int i = 0; i < F_PTS; ++i) { t0 += r[i] * sCpX[i]; t1 += r[i] * sCpY[i]; }
    }
    sT[t][0] = t0; sT[t][1] = t1;
  }
#if __has_builtin(__builtin_amdgcn_s_wait_tensorcnt)
  __builtin_amdgcn_s_wait_tensorcnt(0);   // waves with no TDM issue pass instantly
#endif
  __syncthreads();                        // sT ready AND sPH (TDM) visible to all

  // ---- Phase 2: Pp = P_HAT(256x20) @ T(20x2) via WMMA f32 16x16x4 ----
  const int lane = t & 31;
  const int wv   = t >> 5;
  const int m15  = lane & 15;      // M within tile (A) / N column (B,D)
  const int kh   = lane >> 4;      // K-half selector (A/B VGPR lane split)
#if __has_builtin(__builtin_amdgcn_wmma_f32_16x16x4_f32)
  {
    // Build the 5 B fragments once, branchlessly (identical for both M-tiles).
    // B 4x16 layout: v0 holds rows K0/K2 (lane halves), v1 holds K1/K3.
    // Only N columns 0 (Tx) and 1 (Ty) are non-zero.
    const float bmask = (m15 < 2) ? 1.0f : 0.0f;
    const int   bcol  = m15 & 1;
    v2f bfrag[5];
    #pragma unroll
    for (int s = 0; s < 5; ++s) {
      const int kr = s*4 + kh*2;
      bfrag[s].x = sT[kr + 0][bcol] * bmask;
      bfrag[s].y = sT[kr + 1][bcol] * bmask;
    }
    #pragma unroll
    for (int tt = 0; tt < 2; ++tt) {
      const int rowBase = (2*wv + tt) * 16;
      v8f acc = {0.f, 0.f, 0.f, 0.f, 0.f, 0.f, 0.f, 0.f};
      // A 16x4 f32 layout: lanes 0-15 hold K={k0,k0+1}, lanes 16-31 K={k0+2,k0+3}
      const float* arow = sPH + (rowBase + m15) * PHPAD + kh * 2;
      #pragma unroll
      for (int s = 0; s < 5; ++s) {
        v2f a = *(const v2f*)(arow + s*4);          // ds_load_b64, 8B aligned
        acc = __builtin_amdgcn_wmma_f32_16x16x4_f32(false, a, false, bfrag[s],
                                                    (short)0, acc, false, false);
      }
      // D 16x16 layout: VGPR r -> M=r (lanes 0-15) / M=8+r (lanes 16-31), N=lane&15.
      if (m15 < 2) {
        float* dst = m15 ? sPpy : sPpx;
        const int mb = rowBase + kh * 8;
        #pragma unroll
        for (int r = 0; r < 8; ++r) dst[mb + r] = acc[r];
      }
    }
  }
#else
  { // scalar fallback (host pass / missing builtin): one row per thread
    const float* r = sPH + (size_t)t * PHPAD;
    float px = 0.f, py = 0.f;
    for (int i = 0; i < NCTRL; ++i) { px += r[i] * sT[i][0]; py += r[i] * sT[i][1]; }
    sPpx[t] = px; sPpy[t] = py;
  }
#endif
  __syncthreads();

  // ---- Phase 3: per-pixel sampling state (registers), then channel loop ----
  const float x1f = sMeta[0], y1f = sMeta[1], wc = sMeta[2], hc = sMeta[3];
  const float gx = fminf(fmaxf(((sPpx[t] + 1.0f) * wc - 1.0f) * 0.5f, 0.0f), wc - 1.0f) + x1f;
  const float gy = fminf(fmaxf(((sPpy[t] + 1.0f) * hc - 1.0f) * 0.5f, 0.0f), hc - 1.0f) + y1f;
  const float x0 = floorf(gx), y0 = floorf(gy);
  const float wx = gx - x0,   wy = gy - y0;
  int x0i = (int)x0; x0i = x0i < 0 ? 0 : (x0i > WD-1 ? WD-1 : x0i);
  int y0i = (int)y0; y0i = y0i < 0 ? 0 : (y0i > HD-1 ? HD-1 : y0i);
  const int x1i = x0i + 1 > WD-1 ? WD-1 : x0i + 1;
  const int y1i = y0i + 1 > HD-1 ? HD-1 : y0i + 1;
  const int o00 = y0i*WD + x0i, o01 = y0i*WD + x1i;
  const int o10 = y1i*WD + x0i, o11 = y1i*WD + x1i;
  const float w00 = (1.f-wx)*(1.f-wy), w01 = wx*(1.f-wy);
  const float w10 = (1.f-wx)*wy,       w11 = wx*wy;

  const int bi = bidx[k];
  const int c0 = blockIdx.y * CH_PER;
  const float* fb = feats + (size_t)(bi * CD + c0) * HWD;
  float* ob = out + ((size_t)k * CD + c0) * NPTS + t;

  #pragma unroll 4
  for (int c = 0; c < CH_PER; ++c) {
    const float* fc = fb + (size_t)c * HWD;
    float v = fc[o00]*w00 + fc[o01]*w01 + fc[o10]*w10 + fc[o11]*w11;
    ob[(size_t)c * NPTS] = v;
  }
}

// ---------------------------------------------------------------------------
extern "C" void kernel_launch(void* const* d_in, const int* in_sizes, int n_in,
                              void* d_out, int out_size, void* d_ws, size_t ws_size,
                              hipStream_t stream) {
  const float* feats  = (const float*)d_in[0];
  const float* points = (const float*)d_in[1];
  const int*   bidx   = (const int*)d_in[2];
  float*       out    = (float*)d_out;
  const int K = in_sizes[2];                 // batch_idx element count

  float* inv_dc = (float*)d_ws;              // 289 floats (rounded region: 320)
  float* p_hat  = inv_dc + 320;              // 256*20 floats; total ~21.8 KB

  hipLaunchKernelGGL(tps_setup_kernel, dim3(1), dim3(256), 0, stream,
                     inv_dc, p_hat);
  hipLaunchKernelGGL(tps_main_kernel, dim3(K, CH_SPLIT), dim3(256), 0, stream,
                     feats, points, bidx, inv_dc, p_hat, out);
}